// ModelWithEdgeFeatures_conv_74517682586171
// MI455X (gfx1250) — compile-verified
//
#include <hip/hip_runtime.h>

typedef float v2f __attribute__((ext_vector_type(2)));
typedef float v8f __attribute__((ext_vector_type(8)));

constexpr int   N_NODES   = 200000;
constexpr int   N_EDGES   = 600000;
constexpr int   NUM_G     = 8192;
constexpr int   IN_CH     = 32;
constexpr int   H0        = 128;
constexpr int   H1        = 256;
constexpr int   MLP_H     = 256;
constexpr int   N_CLS     = 10;
constexpr float MAX_SIZE  = 40.0f;
constexpr float BN_EPS    = 1e-5f;

// ---------------- utility: vectorized zero ----------------
__global__ void zero4_kernel(float4* __restrict__ p, int n4) {
    int i = blockIdx.x * blockDim.x + threadIdx.x;
    if (i < n4) p[i] = make_float4(0.f, 0.f, 0.f, 0.f);
}

// ---------------- degree / norm precompute ----------------
__global__ void deg_kernel(const int* __restrict__ dst, const int* __restrict__ et,
                           float* __restrict__ deg) {
    int e = blockIdx.x * blockDim.x + threadIdx.x;
    if (e >= N_EDGES) return;
    atomicAdd(&deg[et[e] * N_NODES + dst[e]], 1.0f);
}

__global__ void dinv_kernel(float* __restrict__ deg, int n) {
    int i = blockIdx.x * blockDim.x + threadIdx.x;
    if (i >= n) return;
    float d = deg[i];
    deg[i] = (d > 0.f) ? rsqrtf(d) : 0.f;   // in place: deg -> dinv
}

__global__ void norm_kernel(const int* __restrict__ src, const int* __restrict__ dst,
                            const int* __restrict__ et, const float* __restrict__ dinv,
                            float* __restrict__ nrm) {
    int e = blockIdx.x * blockDim.x + threadIdx.x;
    if (e >= N_EDGES) return;
    int t = et[e];
    nrm[e] = dinv[t * N_NODES + src[e]] * dinv[t * N_NODES + dst[e]];
}

// ---------------- edge scatter in INPUT feature space ----------------
// One WAVE per edge: metadata loads are wave-uniform (single request),
// lanes stride the K channels with coalesced reads.
__global__ void scatter_kernel(const int* __restrict__ src, const int* __restrict__ dst,
                               const int* __restrict__ et, const float* __restrict__ nrm,
                               const float* __restrict__ x, float* __restrict__ agg,
                               int rel, int K) {
    int e    = (blockIdx.x * blockDim.x + threadIdx.x) >> 5;   // wave index == edge
    int lane = threadIdx.x & 31;
    if (e >= N_EDGES) return;
    if (et[e] != rel) return;                                   // wave-uniform branch
    float nm = nrm[e];
    const float* xs = x   + (long)src[e] * K;
    float*       ad = agg + (long)dst[e] * K;
    for (int c = lane; c < K; c += 32)
        atomicAdd(&ad[c], xs[c] * nm);
}

// ---------------- WMMA f32 GEMM, 64x32 tile per wave, fused epilogue ----------------
// out[M,H] (+)= relu(A[M,K] @ W[K,H] + bias + rvec[row]*rscale*rrow[col])
// 4 M-subtiles x 2 N-subtiles of 16x16: 8 loads per 8 wmma in steady state.
// M multiple of 64, H multiple of 32, K multiple of 4.
__global__ __launch_bounds__(256)
void wmma_gemm_relu(const float* __restrict__ A, const float* __restrict__ W,
                    const float* __restrict__ bias, float* __restrict__ out,
                    int M, int K, int H, int accumulate,
                    const float* __restrict__ rvec, const float* __restrict__ rrow,
                    float rscale) {
    int lane = threadIdx.x & 31;
    int wave = threadIdx.x >> 5;
    int tile = blockIdx.x * 8 + wave;
    int ntiles = H >> 5;                       // 32-wide N blocks
    int total  = (M >> 6) * ntiles;            // 64-tall M blocks
    if (tile >= total) return;                 // wave-uniform: EXEC stays all-1
    int m0 = (tile / ntiles) << 6;
    int n0 = (tile - (tile / ntiles) * ntiles) << 5;

    int lh = lane & 15;
    int ks = (lane >> 4) << 1;                 // lanes 0-15 -> K{0,1}; 16-31 -> K{2,3}
    const float* a0 = A + (long)(m0 + lh) * K + ks;
    const float* b0 = W + n0 + lh;

    v8f acc[4][2] = {};
    for (int kb = 0; kb < K; kb += 4) {
        v2f a[4], b[2];
#pragma unroll
        for (int i = 0; i < 4; ++i)            // A fragments: 8B-aligned b64 loads
            a[i] = *reinterpret_cast<const v2f*>(a0 + (long)(i * 16) * K + kb);
#pragma unroll
        for (int j = 0; j < 2; ++j) {          // B fragments: rows kb+ks, kb+ks+1
            b[j].x = b0[(long)(kb + ks) * H + j * 16];
            b[j].y = b0[(long)(kb + ks + 1) * H + j * 16];
        }
#pragma unroll
        for (int i = 0; i < 4; ++i)
#pragma unroll
            for (int j = 0; j < 2; ++j)
                acc[i][j] = __builtin_amdgcn_wmma_f32_16x16x4_f32(
                    false, a[i], false, b[j], (short)0, acc[i][j], false, false);
    }

    int rowoff = (lane >> 4) << 3;             // VGPR r -> row base + r (+8 for hi half)
#pragma unroll
    for (int j = 0; j < 2; ++j) {
        int col = n0 + j * 16 + lh;
        float bb = bias[col];
        float rr = (rrow != nullptr) ? rrow[col] * rscale : 0.f;
#pragma unroll
        for (int i = 0; i < 4; ++i) {
#pragma unroll
            for (int r = 0; r < 8; ++r) {
                int row = m0 + i * 16 + rowoff + r;
                float v = acc[i][j][r] + bb;
                if (rvec != nullptr) v += rvec[row] * rr;
                v = fmaxf(v, 0.f);
                float* o = &out[(long)row * H + col];
                if (accumulate) *o += v; else *o = v;   // tile owned by wave: no atomics
            }
        }
    }
}

// ---------------- BatchNorm over nodes ----------------
__global__ void bn_stats_kernel(const float* __restrict__ h, float* __restrict__ sums,
                                float* __restrict__ sqs, int M, int rowsPerBlock) {
    int H = blockDim.x;
    int c = threadIdx.x;
    int r0 = blockIdx.x * rowsPerBlock;
    int r1 = min(r0 + rowsPerBlock, M);
    float s = 0.f, q = 0.f;
    for (int r = r0; r < r1; ++r) {
        float v = h[(long)r * H + c];
        s += v; q += v * v;
    }
    atomicAdd(&sums[c], s);
    atomicAdd(&sqs[c], q);
}

__global__ void bn_apply_kernel(float* __restrict__ h, const float* __restrict__ sums,
                                const float* __restrict__ sqs,
                                const float* __restrict__ gamma, const float* __restrict__ beta,
                                int n, int hmask, float invM) {
    int i = blockIdx.x * blockDim.x + threadIdx.x;
    if (i >= n) return;
    int c = i & hmask;
    float mean = sums[c] * invM;
    float var  = sqs[c] * invM - mean * mean;
    h[i] = gamma[c] * (h[i] - mean) * rsqrtf(var + BN_EPS) + beta[c];
}

// ---------------- global_add_pool ----------------
__global__ void pool_kernel(const float* __restrict__ h, const int* __restrict__ batch,
                            float* __restrict__ pooled, float* __restrict__ counts) {
    int node = blockIdx.x;
    int c = threadIdx.x;                        // blockDim == H1 == 256
    int b = batch[node];
    atomicAdd(&pooled[(long)b * H1 + c], h[(long)node * H1 + c]);
    if (c == 0) atomicAdd(&counts[b], 1.0f);
}

// ---------------- final tiny FC2 ----------------
__global__ void fc2_kernel(const float* __restrict__ a, const float* __restrict__ W,
                           const float* __restrict__ b, float* __restrict__ out) {
    int t = threadIdx.x;                        // blockDim == 160 -> 16 graphs/block
    int g = blockIdx.x * 16 + t / N_CLS;
    int c = t - (t / N_CLS) * N_CLS;
    float acc = b[c];
    const float* ag = a + (long)g * MLP_H;
    for (int k = 0; k < MLP_H; ++k) acc += ag[k] * W[k * N_CLS + c];
    out[g * N_CLS + c] = acc;
}

// ---------------- host orchestration ----------------
extern "C" void kernel_launch(void* const* d_in, const int* in_sizes, int n_in,
                              void* d_out, int out_size, void* d_ws, size_t ws_size,
                              hipStream_t stream) {
    const float* x    = (const float*)d_in[0];
    const int*   ei   = (const int*)d_in[1];
    const int*   srcp = ei;
    const int*   dstp = ei + N_EDGES;
    const int*   et   = (const int*)d_in[2];
    const int*   batch= (const int*)d_in[3];
    const float* Wc0  = (const float*)d_in[4];
    const float* bc0  = (const float*)d_in[5];
    const float* Wi0  = (const float*)d_in[6];
    const float* bi0  = (const float*)d_in[7];
    const float* g0   = (const float*)d_in[8];
    const float* be0  = (const float*)d_in[9];
    const float* Wc1  = (const float*)d_in[10];
    const float* bc1  = (const float*)d_in[11];
    const float* Wi1  = (const float*)d_in[12];
    const float* bi1  = (const float*)d_in[13];
    const float* g1   = (const float*)d_in[14];
    const float* be1  = (const float*)d_in[15];
    const float* Wfc1 = (const float*)d_in[16];
    const float* bfc1 = (const float*)d_in[17];
    const float* Wfc2 = (const float*)d_in[18];
    const float* bfc2 = (const float*)d_in[19];
    float* out = (float*)d_out;

    // workspace layout (floats), ~432 MB total
    float* ws     = (float*)d_ws;
    float* dinv   = ws;                                   // 3*N
    float* nrm    = dinv   + 3L * N_NODES;                // E
    float* aggx   = nrm    + N_EDGES;                     // N*128 (reused both layers)
    float* h0     = aggx   + (long)N_NODES * 128;         // N*128
    float* h1     = h0     + (long)N_NODES * H0;          // N*256
    float* bnsum  = h1     + (long)N_NODES * H1;          // 256
    float* bnsq   = bnsum  + 256;                         // 256
    float* pooled = bnsq   + 256;                         // G*256
    float* counts = pooled + (long)NUM_G * H1;            // G
    float* fc1o   = counts + NUM_G;                       // G*256

    auto zero = [&](float* p, long n) {
        int n4 = (int)(n / 4);
        zero4_kernel<<<(n4 + 255) / 256, 256, 0, stream>>>((float4*)p, n4);
    };
    auto gemm = [&](const float* A, const float* W, const float* bias, float* O,
                    int M, int K, int H, int acc, const float* rv, const float* rr, float rs) {
        int waves = (M >> 6) * (H >> 5);                  // one 64x32 tile per wave
        wmma_gemm_relu<<<(waves + 7) / 8, 256, 0, stream>>>(A, W, bias, O, M, K, H, acc, rv, rr, rs);
    };
    auto scatter = [&](const float* X, float* AGG, int rel, int K) {
        long thr = (long)N_EDGES * 32;                    // one wave per edge
        scatter_kernel<<<(int)((thr + 255) / 256), 256, 0, stream>>>(
            srcp, dstp, et, nrm, X, AGG, rel, K);
    };

    // ---- degrees & per-edge norms (shared by both layers) ----
    zero(dinv, 3L * N_NODES);
    deg_kernel<<<(N_EDGES + 255) / 256, 256, 0, stream>>>(dstp, et, dinv);
    dinv_kernel<<<(3 * N_NODES + 255) / 256, 256, 0, stream>>>(dinv, 3 * N_NODES);
    norm_kernel<<<(N_EDGES + 255) / 256, 256, 0, stream>>>(srcp, dstp, et, dinv, nrm);

    // ---- layer 0: IN_CH=32 -> H0=128 ----
    gemm(x, Wi0, bi0, h0, N_NODES, IN_CH, H0, /*acc=*/0, nullptr, nullptr, 0.f);
    for (int t = 0; t < 3; ++t) {
        zero(aggx, (long)N_NODES * IN_CH);
        scatter(x, aggx, t, IN_CH);
        gemm(aggx, Wc0 + (long)t * IN_CH * H0, bc0 + t * H0, h0,
             N_NODES, IN_CH, H0, /*acc=*/1, nullptr, nullptr, 0.f);
    }
    zero(bnsum, 512);
    bn_stats_kernel<<<(N_NODES + 499) / 500, H0, 0, stream>>>(h0, bnsum, bnsq, N_NODES, 500);
    bn_apply_kernel<<<((long)N_NODES * H0 + 255) / 256, 256, 0, stream>>>(
        h0, bnsum, bnsq, g0, be0, N_NODES * H0, H0 - 1, 1.0f / N_NODES);

    // ---- layer 1: H0=128 -> H1=256 ----
    gemm(h0, Wi1, bi1, h1, N_NODES, H0, H1, /*acc=*/0, nullptr, nullptr, 0.f);
    for (int t = 0; t < 3; ++t) {
        zero(aggx, (long)N_NODES * H0);
        scatter(h0, aggx, t, H0);
        gemm(aggx, Wc1 + (long)t * H0 * H1, bc1 + t * H1, h1,
             N_NODES, H0, H1, /*acc=*/1, nullptr, nullptr, 0.f);
    }
    zero(bnsum, 512);
    bn_stats_kernel<<<(N_NODES + 499) / 500, H1, 0, stream>>>(h1, bnsum, bnsq, N_NODES, 500);
    bn_apply_kernel<<<((long)N_NODES * H1 + 255) / 256, 256, 0, stream>>>(
        h1, bnsum, bnsq, g1, be1, N_NODES * H1, H1 - 1, 1.0f / N_NODES);

    // ---- pooling ----
    zero(pooled, (long)NUM_G * H1 + NUM_G);               // pooled + counts (contiguous)
    pool_kernel<<<N_NODES, H1, 0, stream>>>(h1, batch, pooled, counts);

    // ---- FC1: [G,256]@[256,256] + rank-1 (counts/40 x Wfc1[256,:]) + bias, relu ----
    gemm(pooled, Wfc1, bfc1, fc1o, NUM_G, MLP_H, MLP_H, /*acc=*/0,
         counts, Wfc1 + (long)MLP_H * MLP_H, 1.0f / MAX_SIZE);

    // ---- FC2 ----
    fc2_kernel<<<NUM_G / 16, 160, 0, stream>>>(fc1o, Wfc2, bfc2, out);
}